// KnowldgeShifter_61546881351881
// MI455X (gfx1250) — compile-verified
//
#include <hip/hip_runtime.h>

#define NEGINF -1.0e20f

typedef __attribute__((ext_vector_type(2))) float v2f;
typedef __attribute__((ext_vector_type(8))) float v8f;

constexpr int NB = 32;   // batch
constexpr int KP = 32;   // pool entries
constexpr int TT = 128;  // tokens
constexpr int HH = 768;  // hidden
constexpr int H2 = 1536; // 2*H

// workspace layout (floats)
constexpr int WS_CQK    = 0;                  // 32*1536
constexpr int WS_CQKPRO = NB * H2;            // 32*768
constexpr int WS_V      = WS_CQKPRO + NB*HH;  // 32*768

// ---------------- build cqk = concat(ctx[:,2,:], tracked) ----------------
__global__ void build_cqk_kernel(const float* __restrict__ ctx,
                                 const float* __restrict__ tracked,
                                 float* __restrict__ cqk) {
    int i = blockIdx.x * blockDim.x + threadIdx.x;
    if (i >= NB * H2) return;
    int n = i / H2, j = i - n * H2;
    cqk[i] = (j < HH) ? ctx[(n * 3 + 2) * HH + j] : tracked[n * HH + (j - HH)];
}

// ---------------- cqk_pro = cqk(32x1536) @ W_cqk^T + b_cqk ----------------
// One wave per 16x16 tile; tiles = 2 x 48 = 96 waves. v_wmma_f32_16x16x4_f32.
__global__ void gemm_cqkpro_kernel(const float* __restrict__ A,     // 32x1536
                                   const float* __restrict__ Wcqk,  // 768x1536 (row-major)
                                   const float* __restrict__ bias,  // 768
                                   float* __restrict__ Cout) {      // 32x768
    int wave = (blockIdx.x * blockDim.x + threadIdx.x) >> 5;
    int lane = threadIdx.x & 31;
    if (wave >= 2 * (HH / 16)) return;
    int tm = wave & 1;          // 0..1
    int tn = wave >> 1;         // 0..47
    int m  = tm * 16 + (lane & 15);
    int nc = tn * 16 + (lane & 15);
    int kOff = (lane >> 4) << 1;   // 0 or 2

    const float* Arow = A + m * H2;
    const float* Wrow = Wcqk + nc * H2;  // B_mat[k][nc] = W_cqk[nc][k]
    v8f acc = {};
    for (int k0 = 0; k0 < H2; k0 += 4) {
        v2f a, b;
        a[0] = Arow[k0 + kOff];     a[1] = Arow[k0 + kOff + 1];
        b[0] = Wrow[k0 + kOff];     b[1] = Wrow[k0 + kOff + 1];
        acc = __builtin_amdgcn_wmma_f32_16x16x4_f32(false, a, false, b,
                                                    (short)0, acc, false, false);
    }
    float bv = bias[nc];
    int rbase = tm * 16 + ((lane >> 4) << 3);  // D: VGPR i -> row i (+8 for hi lanes)
#pragma unroll
    for (int i = 0; i < 8; ++i)
        Cout[(rbase + i) * HH + nc] = acc[i] + bv;
}

// ---------------- v = cqk_pro(32x768) @ W_k(768x768, row-major) ----------------
__global__ void gemm_v_kernel(const float* __restrict__ A,   // 32x768
                              const float* __restrict__ Wk,  // 768x768, B_mat[k][n]=Wk[k*768+n]
                              float* __restrict__ Vout) {    // 32x768
    int wave = (blockIdx.x * blockDim.x + threadIdx.x) >> 5;
    int lane = threadIdx.x & 31;
    if (wave >= 2 * (HH / 16)) return;
    int tm = wave & 1;
    int tn = wave >> 1;
    int m  = tm * 16 + (lane & 15);
    int nc = tn * 16 + (lane & 15);
    int kOff = (lane >> 4) << 1;

    const float* Arow = A + m * HH;
    v8f acc = {};
    for (int k0 = 0; k0 < HH; k0 += 4) {
        v2f a, b;
        a[0] = Arow[k0 + kOff];             a[1] = Arow[k0 + kOff + 1];
        b[0] = Wk[(k0 + kOff) * HH + nc];   b[1] = Wk[(k0 + kOff + 1) * HH + nc];
        acc = __builtin_amdgcn_wmma_f32_16x16x4_f32(false, a, false, b,
                                                    (short)0, acc, false, false);
    }
    int rbase = tm * 16 + ((lane >> 4) << 3);
#pragma unroll
    for (int i = 0; i < 8; ++i)
        Vout[(rbase + i) * HH + nc] = acc[i];
}

// ---------------- score[n,k] = <pe1[n,k], v[n]> + <b_k, cqk_pro[n]>, masked ----
__global__ void score_kernel(const float* __restrict__ pe1,      // 32x32x768
                             const float* __restrict__ v,        // 32x768
                             const float* __restrict__ cqk_pro,  // 32x768
                             const float* __restrict__ bk,       // 768
                             const unsigned char* __restrict__ ck_mask, // 32x32 bool
                             float* __restrict__ score) {        // 32x32
    int wave = (blockIdx.x * blockDim.x + threadIdx.x) >> 5;
    int lane = threadIdx.x & 31;
    if (wave >= NB * KP) return;
    int n = wave >> 5;
    int k = wave & 31;
    const float* a  = pe1 + (size_t)(n * KP + k) * HH;
    const float* vv = v + n * HH;
    const float* cp = cqk_pro + n * HH;
    float s = 0.f;
    for (int h = lane; h < HH; h += 32)
        s = fmaf(a[h], vv[h], fmaf(bk[h], cp[h], s));
#pragma unroll
    for (int off = 16; off > 0; off >>= 1)
        s += __shfl_xor(s, off, 32);
    if (lane == 0)
        score[n * KP + k] = ck_mask[n * KP + k] ? s : NEGINF;
}

// ---------------- shifted_encoded gather: 12.6 MB float4 copy ----------------
__global__ void gather_encoded_kernel(const float4* __restrict__ pe0,
                                      const int* __restrict__ label,
                                      float4* __restrict__ out) {
    const int perN = TT * HH / 4;  // 24576 float4 per n
    const int total = NB * perN;
    for (int i = blockIdx.x * blockDim.x + threadIdx.x; i < total;
         i += gridDim.x * blockDim.x) {
        int n = i / perN;
        int rem = i - n * perN;
        out[i] = pe0[(size_t)(n * KP + label[n]) * perN + rem];
    }
}

// ---------------- small gathers: mask->float, tokens->float, use copy --------
__global__ void gather_small_kernel(const unsigned char* __restrict__ pmask,
                                    const int* __restrict__ ptok,
                                    const float4* __restrict__ pe1,
                                    const int* __restrict__ label,
                                    float* __restrict__ out_mask,
                                    float* __restrict__ out_index,
                                    float4* __restrict__ out_use) {
    int i = blockIdx.x * blockDim.x + threadIdx.x;
    if (i < NB * TT) {  // 4096
        int n = i / TT, t = i - n * TT;
        int src = (n * KP + label[n]) * TT + t;
        out_mask[i]  = pmask[src] ? 1.0f : 0.0f;
        out_index[i] = (float)ptok[src];
    }
    const int perN = HH / 4;  // 192 float4 per n
    if (i < NB * perN) {  // 6144
        int n = i / perN;
        int rem = i - n * perN;
        out_use[i] = pe1[(size_t)(n * KP + label[n]) * perN + rem];
    }
}

extern "C" void kernel_launch(void* const* d_in, const int* in_sizes, int n_in,
                              void* d_out, int out_size, void* d_ws, size_t ws_size,
                              hipStream_t stream) {
    const float* ctx      = (const float*)d_in[0];   // (32,3,768)
    const float* tracked  = (const float*)d_in[1];   // (32,768)
    const float* pe0      = (const float*)d_in[2];   // (32,32,128,768)
    const float* pe1      = (const float*)d_in[3];   // (32,32,768)
    const unsigned char* pmask  = (const unsigned char*)d_in[4]; // (32,32,128) bool
    const unsigned char* ckmask = (const unsigned char*)d_in[5]; // (32,32) bool
    const int*   label    = (const int*)d_in[6];     // (32,)
    const int*   ptok     = (const int*)d_in[7];     // (32,32,128)
    const float* Wcqk     = (const float*)d_in[8];   // (768,1536)
    const float* bcqk     = (const float*)d_in[9];   // (768,)
    const float* Wk       = (const float*)d_in[10];  // (768,768)
    const float* bk       = (const float*)d_in[11];  // (768,)

    float* ws      = (float*)d_ws;
    float* cqk     = ws + WS_CQK;
    float* cqk_pro = ws + WS_CQKPRO;
    float* v       = ws + WS_V;

    // output layout (flat f32, return order)
    float* out      = (float*)d_out;
    float* o_score  = out;                         // 1024
    float* o_enc    = out + NB * KP;               // 3145728
    float* o_mask   = o_enc + (size_t)NB * TT * HH;// 4096
    float* o_use    = o_mask + NB * TT;            // 24576
    float* o_idx    = o_use + NB * HH;             // 4096

    // 1) build concatenated query (32x1536)
    build_cqk_kernel<<<(NB * H2 + 255) / 256, 256, 0, stream>>>(ctx, tracked, cqk);

    // 2) cqk_pro = cqk @ W_cqk^T + b_cqk   (96 waves -> 12 blocks x 8 waves)
    gemm_cqkpro_kernel<<<12, 256, 0, stream>>>(cqk, Wcqk, bcqk, cqk_pro);

    // 3) v = cqk_pro @ W_k
    gemm_v_kernel<<<12, 256, 0, stream>>>(cqk_pro, Wk, v);

    // 4) scores: 1024 waves -> 128 blocks x 8 waves
    score_kernel<<<128, 256, 0, stream>>>(pe1, v, cqk_pro, bk, ckmask, o_score);

    // 5) bulk gather of shifted_encoded (float4)
    gather_encoded_kernel<<<3072, 256, 0, stream>>>((const float4*)pe0, label,
                                                    (float4*)o_enc);

    // 6) small gathers
    gather_small_kernel<<<24, 256, 0, stream>>>(pmask, ptok, (const float4*)pe1,
                                                label, o_mask, o_idx, (float4*)o_use);
}